// PointTransformerV3_78357383348686
// MI455X (gfx1250) — compile-verified
//
#include <hip/hip_runtime.h>

// Mathematically-reduced PointTransformer block:
//   softmax over axis j followed by sum over axis j == 1  =>  out = relu((x@wv)@wo + bo)
// Two 768x128x128 fp32 GEMMs on the CDNA5 fp32 WMMA path (V_WMMA_F32_16X16X4_F32).

typedef __attribute__((ext_vector_type(2))) float v2f;
typedef __attribute__((ext_vector_type(8))) float v8f;

// C[M,128] = A[M,128] @ B[128,128]   (optionally + bias, relu)
// One wave -> one 16-row x 64-col output stripe (4 accumulators, A-fragment reuse).
// Block = 256 threads = 8 waves; tasks = (M/16) * 2.
__global__ __launch_bounds__(256) void gemm128_f32_wmma(
    const float* __restrict__ A, const float* __restrict__ B,
    const float* __restrict__ bias, float* __restrict__ C,
    int M, int fuse_bias_relu)
{
    const int lane = threadIdx.x & 31;
    const int wid  = threadIdx.x >> 5;
    const int task = blockIdx.x * 8 + wid;
    const int mt   = task >> 1;         // 16-row tile
    const int ng   = task & 1;          // 64-col group
    const int mbase = mt * 16;
    const int nbase = ng * 64;
    if (mbase >= M) return;             // uniform per wave

    // fp32 WMMA 16x16x4 lane layout (ISA 7.12.2):
    //   A 16x4: lanes 0-15 hold K={0,1} in v0/v1, lanes 16-31 hold K={2,3}
    //   B 4x16: lanes 0-15 hold K={0,1} rows at N=lane, lanes 16-31 hold K={2,3}
    //   C 16x16: vgpr i -> M = i (lanes 0-15) or 8+i (lanes 16-31), N = lane%16
    const int half = (lane < 16) ? 0 : 2;
    const int lrow = lane & 15;

    v8f acc0 = {}, acc1 = {}, acc2 = {}, acc3 = {};

    const float* arowp = A + (size_t)(mbase + lrow) * 128;

    for (int k0 = 0; k0 < 128; k0 += 4) {
        const int kr = k0 + half;
        v2f a = *(const v2f*)(arowp + kr);          // A[m][kr], A[m][kr+1]

        const float* b0p = B + (size_t)kr * 128 + nbase + lrow;   // K=kr   row
        const float* b1p = b0p + 128;                             // K=kr+1 row
        v2f b;

        b.x = b0p[0];  b.y = b1p[0];
        acc0 = __builtin_amdgcn_wmma_f32_16x16x4_f32(
                   false, a, false, b, (short)0, acc0, false, false);
        b.x = b0p[16]; b.y = b1p[16];
        acc1 = __builtin_amdgcn_wmma_f32_16x16x4_f32(
                   false, a, false, b, (short)0, acc1, false, false);
        b.x = b0p[32]; b.y = b1p[32];
        acc2 = __builtin_amdgcn_wmma_f32_16x16x4_f32(
                   false, a, false, b, (short)0, acc2, false, false);
        b.x = b0p[48]; b.y = b1p[48];
        acc3 = __builtin_amdgcn_wmma_f32_16x16x4_f32(
                   false, a, false, b, (short)0, acc3, false, false);
    }

    const int rbase = mbase + ((lane < 16) ? 0 : 8);
    v8f accs[4] = {acc0, acc1, acc2, acc3};

#pragma unroll
    for (int t = 0; t < 4; ++t) {
        const int ncol = nbase + t * 16 + lrow;
        const float bval = fuse_bias_relu ? bias[ncol] : 0.0f;
#pragma unroll
        for (int i = 0; i < 8; ++i) {
            float v = accs[t][i] + bval;
            if (fuse_bias_relu) v = fmaxf(v, 0.0f);
            C[(size_t)(rbase + i) * 128 + ncol] = v;
        }
    }
}

extern "C" void kernel_launch(void* const* d_in, const int* in_sizes, int n_in,
                              void* d_out, int out_size, void* d_ws, size_t ws_size,
                              hipStream_t stream) {
    (void)in_sizes; (void)n_in; (void)out_size; (void)ws_size;
    // setup_inputs() order: x pos pe_w1 pe_b1 pe_w2 pe_b2 am_w1 am_b1 am_w2 am_b2 wq wk wv wo bo
    const float* x  = (const float*)d_in[0];
    const float* wv = (const float*)d_in[12];
    const float* wo = (const float*)d_in[13];
    const float* bo = (const float*)d_in[14];
    float* out  = (float*)d_out;
    float* vtmp = (float*)d_ws;          // 768*128 fp32 staging for v = x@wv

    const int M = 768;
    const int tasks = (M / 16) * 2;      // 96 wave-stripes
    dim3 block(256);                     // 8 waves
    dim3 grid(tasks / 8);                // 12 blocks

    // v = x @ wv
    gemm128_f32_wmma<<<grid, block, 0, stream>>>(x, wv, nullptr, vtmp, M, 0);
    // out = relu(v @ wo + bo)
    gemm128_f32_wmma<<<grid, block, 0, stream>>>(vtmp, wo, bo, out, M, 1);
}